// BeamSearchVarLen_87179246174259
// MI455X (gfx1250) — compile-verified
//
#include <hip/hip_runtime.h>
#include <cstdint>

// Fused KV-cache beam reorder + suffix append for MI455X (gfx1250).
// Pure bandwidth problem (~1 GB traffic, 0 FLOPs) -> CDNA5 async global<->LDS
// DMA path (ASYNCcnt) for a register-free streaming copy, with non-temporal
// stores so the 512 MB write stream does not evict the gather-reuse lines
// (duplicate beam ancestors) from the 192 MB L2.

#define LAYERS 8
#define BEAMS  128
#define HEADS  8
#define TLEN   128
#define HDIM   64

#define TILE_F4        (TLEN * HDIM / 4)                       // 2048 float4 per (l,g,h)
#define PER_TENSOR_F4  (LAYERS * BEAMS * HEADS * TILE_F4)      // 16,777,216 float4

typedef __attribute__((ext_vector_type(4))) float v4f;  // native 16B vector

// ---------------------------------------------------------------------------
// Kernel 1: gather-copy whole (l,g,h) tiles, staged through LDS with async DMA.
// One block per (kv, l, g, h): 32 KB tile, 256 threads x 8 b128 chunks each.
// Each thread stores exactly the chunks it loaded -> no barrier needed, only
// the per-wave s_wait_asynccnt between the load burst and the store burst.
// Loads: default RT policy (beam-gather reuse can hit L2).
// Stores: TH_STORE_NT (write-once stream, keep it out of L2).
// ---------------------------------------------------------------------------
__global__ __launch_bounds__(256) void kv_gather_copy(
    const float* __restrict__ kbuf, const float* __restrict__ vbuf,
    const int* __restrict__ beam, float* __restrict__ out)
{
    __shared__ v4f stage[TILE_F4];  // 32 KB of the 320 KB/WGP -> 10 blocks/WGP

    const unsigned tile = blockIdx.x;              // [0, 2*L*G*NH)
    const unsigned h  = tile & (HEADS - 1);
    const unsigned g  = (tile >> 3) & (BEAMS - 1);
    const unsigned l  = (tile >> 10) & (LAYERS - 1);
    const unsigned kv = tile >> 13;

    const int sg = beam[g];  // beam ancestry gather index

    const v4f* __restrict__ src =
        reinterpret_cast<const v4f*>(kv ? vbuf : kbuf) +
        (size_t)((l * BEAMS + (unsigned)sg) * HEADS + h) * TILE_F4;
    v4f* __restrict__ dst =
        reinterpret_cast<v4f*>(out) + (size_t)kv * PER_TENSOR_F4 +
        (size_t)((l * BEAMS + g) * HEADS + h) * TILE_F4;

    const unsigned tid = threadIdx.x;

    // Burst of async global->LDS b128 loads (16 B/lane, 512 B per wave-inst).
#pragma unroll
    for (int j = 0; j < 8; ++j) {
        const unsigned c       = tid + j * 256;
        const unsigned lds_off = (unsigned)(uintptr_t)(&stage[c]);  // low 32b = LDS byte addr
        const v4f*     gsrc    = src + c;
        asm volatile("global_load_async_to_lds_b128 %0, %1, off"
                     :: "v"(lds_off), "v"(gsrc)
                     : "memory");
    }
    // Wait for this wave's async loads to land in LDS.
    asm volatile("s_wait_asynccnt 0x0" ::: "memory");

    // Burst of async LDS->global b128 stores of the same chunks, non-temporal.
#pragma unroll
    for (int j = 0; j < 8; ++j) {
        const unsigned c       = tid + j * 256;
        const unsigned lds_off = (unsigned)(uintptr_t)(&stage[c]);
        v4f*           gdst    = dst + c;
        asm volatile("global_store_async_from_lds_b128 %0, %1, off th:TH_STORE_NT"
                     :: "v"(gdst), "v"(lds_off)
                     : "memory");
    }
    // Drain before end (S_ENDPGM also implies wait-idle; this is explicit).
    asm volatile("s_wait_asynccnt 0x0" ::: "memory");
}

// ---------------------------------------------------------------------------
// Kernel 2: overwrite out[kv, l, g, h, pos, :] with k_new/v_new (applied at
// beam g directly, matching the reference's post-gather dynamic_update_slice).
// Runs stream-ordered after kernel 1, so there is no async-store vs vmem-store
// ordering hazard on the pos row. ~4 MB of traffic -> negligible.
// ---------------------------------------------------------------------------
__global__ __launch_bounds__(256) void kv_append_new(
    const float* __restrict__ knew, const float* __restrict__ vnew,
    const int* __restrict__ pos_p, float* __restrict__ out)
{
    const unsigned i = blockIdx.x * 256u + threadIdx.x;  // [0, 2*L*G*NH*16)
    const unsigned hd4 = i & 15;                 // HDIM/4 = 16 float4 per row
    const unsigned h   = (i >> 4) & (HEADS - 1);
    const unsigned g   = (i >> 7) & (BEAMS - 1);
    const unsigned l   = (i >> 14) & (LAYERS - 1);
    const unsigned kv  = i >> 17;

    const int pos = pos_p[0];

    const v4f val = reinterpret_cast<const v4f*>(kv ? vnew : knew)
        [(size_t)((l * BEAMS + g) * HEADS + h) * (HDIM / 4) + hd4];

    v4f* dstp = reinterpret_cast<v4f*>(out) +
        (size_t)kv * PER_TENSOR_F4 +
        ((size_t)((l * BEAMS + g) * HEADS + h) * TLEN + (unsigned)pos) * (HDIM / 4) +
        hd4;
    __builtin_nontemporal_store(val, dstp);
}

// ---------------------------------------------------------------------------
// Host-side launch. Inputs (setup_inputs order):
//   d_in[0] k_buf  [L,G,NH,T,HD] f32     d_in[3] v_new [L,G,NH,1,HD] f32
//   d_in[1] v_buf  [L,G,NH,T,HD] f32     d_in[4] new_beam_idx [G] i32
//   d_in[2] k_new  [L,G,NH,1,HD] f32     d_in[5] pos (scalar) i32
// d_out = concat(k_out, v_out) flat, f32.
// ---------------------------------------------------------------------------
extern "C" void kernel_launch(void* const* d_in, const int* in_sizes, int n_in,
                              void* d_out, int out_size, void* d_ws, size_t ws_size,
                              hipStream_t stream) {
    (void)in_sizes; (void)n_in; (void)out_size; (void)d_ws; (void)ws_size;

    const float* kbuf = (const float*)d_in[0];
    const float* vbuf = (const float*)d_in[1];
    const float* knew = (const float*)d_in[2];
    const float* vnew = (const float*)d_in[3];
    const int*   beam = (const int*)d_in[4];
    const int*   pos  = (const int*)d_in[5];
    float*       out  = (float*)d_out;

    // 2 * L * G * NH = 16384 tiles of 32 KB each.
    kv_gather_copy<<<2 * LAYERS * BEAMS * HEADS, 256, 0, stream>>>(kbuf, vbuf, beam, out);

    // 2 * L * G * NH * (HD/4) = 262144 float4 writes -> 1024 blocks.
    kv_append_new<<<(2 * LAYERS * BEAMS * HEADS * (HDIM / 4)) / 256, 256, 0, stream>>>(
        knew, vnew, pos, out);
}